// GNN_17566416240733
// MI455X (gfx1250) — compile-verified
//
#include <hip/hip_runtime.h>
#include <cstddef>

typedef __attribute__((ext_vector_type(2))) float v2f;
typedef __attribute__((ext_vector_type(8))) float v8f;
typedef unsigned int u32x4 __attribute__((ext_vector_type(4)));
typedef int i32x8 __attribute__((ext_vector_type(8)));
typedef int i32x4 __attribute__((ext_vector_type(4)));

#define HDIM 256
#define EPS 1e-5f

#if defined(__has_builtin)
#if __has_builtin(__builtin_amdgcn_tensor_load_to_lds)
#define HAVE_TDM 1
#else
#define HAVE_TDM 0
#endif
#else
#define HAVE_TDM 0
#endif

// ---------------- degree / normalization ----------------
__global__ void k_init_deg(float* __restrict__ deg, int n) {
    int i = blockIdx.x * blockDim.x + threadIdx.x;
    if (i < n) deg[i] = 1.0f;  // self loop contributes 1
}

__global__ void k_count_deg(const int* __restrict__ col, float* __restrict__ deg, int E) {
    int e = blockIdx.x * blockDim.x + threadIdx.x;
    if (e < E) atomicAdd(&deg[col[e]], 1.0f);
}

__global__ void k_rsqrt(float* __restrict__ deg, int n) {
    int i = blockIdx.x * blockDim.x + threadIdx.x;
    if (i < n) deg[i] = rsqrtf(deg[i]);   // deg >= 1 always (self loop)
}

// ---------------- f32 WMMA GEMM: Y[n,Hc] = X[n,Kd] @ W[Kd,Hc] (+bias,relu) ----
// 8 waves per block, each computing one 16x16 tile along M; all share one
// B tile (Kd x 16, row stride Hc) staged into LDS by the Tensor Data Mover.
template <bool BIAS_RELU>
__global__ void k_gemm_wmma(const float* __restrict__ X, const float* __restrict__ W,
                            const float* __restrict__ bias, float* __restrict__ Y,
                            int nRows, int Kd, int Hc) {
    __shared__ float bs[256 * 16];            // B tile [Kd][16], Kd <= 256 -> 16KB

    const int lane = threadIdx.x;             // 0..31
    const int wave = threadIdx.y;             // 0..7
    int mTile = blockIdx.x * 8 + wave;        // wave-uniform
    const int nTile = blockIdx.y;
    const bool active = (mTile * 16) < nRows;
    if (!active) mTile = 0;                   // clamp: keep EXEC all-1s for WMMA/barrier

    const int l15 = lane & 15;
    const int hi  = lane >> 4;                // 0 or 1

#if HAVE_TDM
    if (wave == 0) {
        // Tensor DMA descriptor (ISA Ch.8): tensor = W[Kd][Hc] f32, tile = 16 cols x Kd rows
        const unsigned long long gaddr = (unsigned long long)(W + (size_t)nTile * 16);
        const unsigned ldsb = (unsigned)(unsigned long long)(void*)&bs[0];
        u32x4 g0;
        g0[0] = 1u;                                           // count=1, no gather
        g0[1] = ldsb;                                         // lds_addr [63:32]
        g0[2] = (unsigned)(gaddr & 0xffffffffu);              // global_addr[31:0]
        g0[3] = (unsigned)((gaddr >> 32) & 0x1ffffffu)        // global_addr[56:32]
              | (2u << 30);                                   // type=2 ("image")
        i32x8 g1;
        g1[0] = (int)(2u << 16);                              // workgroup_mask=0, data_size=4B
        g1[1] = (int)((unsigned)(Hc & 0xffff) << 16);         // tensor_dim0[15:0]
        g1[2] = (int)(((unsigned)Hc >> 16)
              | ((unsigned)(Kd & 0xffff) << 16));             // dim0[31:16] | dim1[15:0]
        g1[3] = (int)(((unsigned)Kd >> 16) | (16u << 16));    // dim1[31:16] | tile_dim0=16
        g1[4] = (int)((unsigned)Kd & 0xffffu);                // tile_dim1=Kd, tile_dim2=0
        g1[5] = (int)(unsigned)Hc;                            // tensor_dim0_stride[31:0]
        g1[6] = 0;                                            // stride0[47:32] | stride1 lo
        g1[7] = 0;
        i32x4 gz4 = {};                                       // groups 2/3 unused (2D tile)
        i32x8 gz8 = {};
        __builtin_amdgcn_tensor_load_to_lds(g0, g1, gz4, gz4, gz8, 0);
        __builtin_amdgcn_s_wait_tensorcnt(0);                 // DMA -> LDS complete
    }
    __syncthreads();
#else
    // fallback: cooperative direct fill of the B tile
    for (int idx = threadIdx.y * 32 + threadIdx.x; idx < Kd * 16; idx += 8 * 32)
        bs[idx] = W[(size_t)(idx >> 4) * Hc + nTile * 16 + (idx & 15)];
    __syncthreads();
#endif

    const int row = mTile * 16 + l15;                  // A row for this lane
    const float* xp = X + (size_t)row * Kd + hi * 2;   // K = hi*2, hi*2+1
    const float* bp = bs + (hi * 2) * 16 + l15;        // B[k+hi*2][l15] in LDS

    v8f acc = {};
    for (int k = 0; k < Kd; k += 4) {
        const float2 av = *(const float2*)xp;          // contiguous -> global b64 load
        v2f a, b;
        a[0] = av.x;  a[1] = av.y;                     // A[M=row][K..K+1]
        b[0] = bp[0]; b[1] = bp[16];                   // B[K][col], B[K+1][col] from LDS
        acc = __builtin_amdgcn_wmma_f32_16x16x4_f32(
            /*neg_a=*/false, a, /*neg_b=*/false, b,
            /*c_mod=*/(short)0, acc, /*reuse_a=*/false, /*reuse_b=*/false);
        xp += 4;
        bp += 64;                                      // 4 rows * 16 floats
    }

    if (active) {
        const int col = nTile * 16 + l15;
        float bval = 0.0f;
        if (BIAS_RELU) bval = bias[col];
        const int mBase = mTile * 16 + hi * 8;         // lanes 16-31 hold rows M+8
#pragma unroll
        for (int v = 0; v < 8; ++v) {
            float val = acc[v];
            if (BIAS_RELU) { val += bval; val = fmaxf(val, 0.0f); }
            Y[(size_t)(mBase + v) * Hc + col] = val;
        }
    }
}

// ---------------- agg = bias + selfloop (xw * dis^2) ----------------
__global__ void k_init_agg(float* __restrict__ agg, const float* __restrict__ bias,
                           const float* __restrict__ xw, const float* __restrict__ dis,
                           int total) {
    int t = blockIdx.x * blockDim.x + threadIdx.x;
    if (t >= total) return;
    float d = dis[t >> 8];
    agg[t] = bias[t & (HDIM - 1)] + xw[t] * d * d;
}

// ---------------- edge scatter: agg[col] += xw[row] * norm ----------------
// 64 threads per edge, float4 per thread, f32 global atomics (HBM-bound path).
__global__ void k_scatter(const int* __restrict__ row, const int* __restrict__ col,
                          const float* __restrict__ dis, const float* __restrict__ xw,
                          float* __restrict__ agg, int E) {
    int t = blockIdx.x * blockDim.x + threadIdx.x;
    int e = t >> 6;
    if (e >= E) return;
    int sub = (t & 63) << 2;
    int r = row[e], c = col[e];
    float w = dis[r] * dis[c];
    const float4 v = *(const float4*)(xw + (size_t)r * HDIM + sub);
    float* dst = agg + (size_t)c * HDIM + sub;
    atomicAdd(dst + 0, v.x * w);
    atomicAdd(dst + 1, v.y * w);
    atomicAdd(dst + 2, v.z * w);
    atomicAdd(dst + 3, v.w * w);
}

// ---------------- LayerNorm + ReLU + residual (wave per node) ----------------
// out = relu(ln(agg)*g+be) * postScale + resCoef * res
__global__ void k_ln_relu(const float* __restrict__ agg, const float* __restrict__ g,
                          const float* __restrict__ be, const float* __restrict__ res,
                          float postScale, float resCoef, float* __restrict__ out, int n) {
    int wid  = threadIdx.x >> 5;
    int lane = threadIdx.x & 31;
    int node = blockIdx.x * 8 + wid;          // 8 waves per 256-thread block
    if (node >= n) return;                    // wave-uniform

    const float* p = agg + (size_t)node * HDIM + lane * 8;
    float4 va = *(const float4*)(p);
    float4 vb = *(const float4*)(p + 4);
    float v[8] = {va.x, va.y, va.z, va.w, vb.x, vb.y, vb.z, vb.w};

    float s = 0.0f, q = 0.0f;
#pragma unroll
    for (int j = 0; j < 8; ++j) { s += v[j]; q += v[j] * v[j]; }
#pragma unroll
    for (int o = 16; o > 0; o >>= 1) {
        s += __shfl_xor(s, o, 32);
        q += __shfl_xor(q, o, 32);
    }
    float mu  = s * (1.0f / HDIM);
    float var = q * (1.0f / HDIM) - mu * mu;
    float rs  = rsqrtf(var + EPS);

    int h = lane * 8;
    const float* rp = res + (size_t)node * HDIM + h;
    float4 ra = *(const float4*)(rp);
    float4 rb = *(const float4*)(rp + 4);
    float r[8] = {ra.x, ra.y, ra.z, ra.w, rb.x, rb.y, rb.z, rb.w};

    float o8[8];
#pragma unroll
    for (int j = 0; j < 8; ++j) {
        float y = (v[j] - mu) * rs * g[h + j] + be[h + j];
        o8[j] = fmaxf(y, 0.0f) * postScale + resCoef * r[j];
    }
    float* op = out + (size_t)node * HDIM + h;
    *(float4*)(op)     = make_float4(o8[0], o8[1], o8[2], o8[3]);
    *(float4*)(op + 4) = make_float4(o8[4], o8[5], o8[6], o8[7]);
}

// ---------------- final projection: out[n] = h[n,128] @ Wf2[128] + bf2 ------
__global__ void k_final(const float* __restrict__ h, const float* __restrict__ w2,
                        const float* __restrict__ b2, float* __restrict__ out, int n) {
    int wid  = threadIdx.x >> 5;
    int lane = threadIdx.x & 31;
    int node = blockIdx.x * 8 + wid;
    if (node >= n) return;
    const float4 a = *(const float4*)(h + (size_t)node * 128 + lane * 4);
    const float4 w = *(const float4*)(w2 + lane * 4);
    float s = a.x * w.x + a.y * w.y + a.z * w.z + a.w * w.w;
#pragma unroll
    for (int o = 16; o > 0; o >>= 1) s += __shfl_xor(s, o, 32);
    if (lane == 0) out[node] = s + b2[0];
}

extern "C" void kernel_launch(void* const* d_in, const int* in_sizes, int n_in,
                              void* d_out, int out_size, void* d_ws, size_t ws_size,
                              hipStream_t stream) {
    const float* x   = (const float*)d_in[0];
    const int*  eidx = (const int*)d_in[1];
    const float* W1  = (const float*)d_in[2];
    const float* b1  = (const float*)d_in[3];
    const float* g1  = (const float*)d_in[4];
    const float* be1 = (const float*)d_in[5];
    const float* W2  = (const float*)d_in[6];
    const float* b2  = (const float*)d_in[7];
    const float* g2  = (const float*)d_in[8];
    const float* be2 = (const float*)d_in[9];
    const float* W3  = (const float*)d_in[10];
    const float* b3  = (const float*)d_in[11];
    const float* g3  = (const float*)d_in[12];
    const float* be3 = (const float*)d_in[13];
    const float* Wf1 = (const float*)d_in[14];
    const float* bf1 = (const float*)d_in[15];
    const float* Wf2 = (const float*)d_in[16];
    const float* bf2 = (const float*)d_in[17];

    const int N = in_sizes[0] / 128;   // 50000
    const int E = in_sizes[1] / 2;     // 800000
    const int* erow = eidx;            // edge_index[0] = source
    const int* ecol = eidx + E;        // edge_index[1] = target

    // workspace layout (floats)
    float* dis = (float*)d_ws;                  // [N]
    float* xw  = dis + N;                       // [N,256]  (also reused for head h[N,128])
    float* agg = xw  + (size_t)N * HDIM;        // [N,256]
    float* xA  = agg + (size_t)N * HDIM;        // x1 then x3
    float* xB  = xA  + (size_t)N * HDIM;        // x2
    (void)ws_size; (void)n_in; (void)out_size;

    // --- symmetric normalization ---
    k_init_deg <<<(N + 255) / 256, 256, 0, stream>>>(dis, N);
    k_count_deg<<<(E + 255) / 256, 256, 0, stream>>>(ecol, dis, E);
    k_rsqrt    <<<(N + 255) / 256, 256, 0, stream>>>(dis, N);

    const dim3 gblk(32, 8);                     // 8 waves share one B tile
    const int  mTiles = (N + 15) / 16;
    const dim3 g256((mTiles + 7) / 8, HDIM / 16);
    const dim3 g128((mTiles + 7) / 8, 128 / 16);
    const int  total     = N * HDIM;
    const int  scatBlks  = (int)(((long long)E * 64 + 255) / 256);
    const int  lnBlks    = (N + 7) / 8;

    // --- conv layer 1: x @ W1, aggregate, LN+ReLU ---
    k_gemm_wmma<false><<<g256, gblk, 0, stream>>>(x, W1, nullptr, xw, N, 128, HDIM);
    k_init_agg <<<(total + 255) / 256, 256, 0, stream>>>(agg, b1, xw, dis, total);
    k_scatter  <<<scatBlks, 256, 0, stream>>>(erow, ecol, dis, xw, agg, E);
    k_ln_relu  <<<lnBlks, 256, 0, stream>>>(agg, g1, be1, agg, 1.0f, 0.0f, xA, N);   // x1

    // --- conv layer 2: x2 = relu(ln(conv(x1))) + 0.7*x1 ---
    k_gemm_wmma<false><<<g256, gblk, 0, stream>>>(xA, W2, nullptr, xw, N, HDIM, HDIM);
    k_init_agg <<<(total + 255) / 256, 256, 0, stream>>>(agg, b2, xw, dis, total);
    k_scatter  <<<scatBlks, 256, 0, stream>>>(erow, ecol, dis, xw, agg, E);
    k_ln_relu  <<<lnBlks, 256, 0, stream>>>(agg, g2, be2, xA, 1.0f, 0.7f, xB, N);    // x2

    // --- conv layer 3: x3 = relu(ln(conv(x2)))*0.7 + x2 ---
    k_gemm_wmma<false><<<g256, gblk, 0, stream>>>(xB, W3, nullptr, xw, N, HDIM, HDIM);
    k_init_agg <<<(total + 255) / 256, 256, 0, stream>>>(agg, b3, xw, dis, total);
    k_scatter  <<<scatBlks, 256, 0, stream>>>(erow, ecol, dis, xw, agg, E);
    k_ln_relu  <<<lnBlks, 256, 0, stream>>>(agg, g3, be3, xB, 0.7f, 1.0f, xA, N);    // x3

    // --- MLP head: h = relu(x3@Wf1+bf1); out = h@Wf2+bf2 ---
    k_gemm_wmma<true><<<g128, gblk, 0, stream>>>(xA, Wf1, bf1, xw, N, HDIM, 128);
    k_final<<<lnBlks, 256, 0, stream>>>(xw, Wf2, bf2, (float*)d_out, N);
}